// SinkhornDistanceFast_8881992368807
// MI455X (gfx1250) — compile-verified
//
#include <hip/hip_runtime.h>
#include <stdint.h>

// Sinkhorn (log-domain) on S[16][2048][2048] fp32, EPS=1, 5 iters.
//   u = -log(N) - LSE over cols of (S + v)   (iters 0,2,4)
//   v = -log(N) - LSE over rows of (S + u)   (iters 1,3), clamp >9e8 -> 0
//   pi = exp(S + u + v)
// Memory-bound: ~1.5GB traffic -> ~65us at 23.3 TB/s.

#define NB   16
#define NN   2048
#define LOG_N 7.624618986159398f   // log(2048)
#define ROW_PAD 4
#define ROW_STRIDE (NN + ROW_PAD)  // 2052: %64==4 -> conflict-free transposed LDS reads

typedef __attribute__((ext_vector_type(2))) float v2f;
typedef __attribute__((ext_vector_type(8))) float v8f;

// ---------------------------------------------------------------------------
// Row-LSE pass. One block = 16 rows of one batch. Stages tile (+v vector) into
// LDS with async-to-LDS loads, computes row max, then row sum-of-exp via
// V_WMMA_F32_16X16X4_F32 (A = 16x4 exp values, B = ones -> D = row sums).
// If WRITE_PI: also emits pi = exp(S + v + u) from the LDS-resident tile.
// ---------------------------------------------------------------------------
template <bool V_ZERO, bool WRITE_PI>
__global__ __launch_bounds__(256) void sinkhorn_row_pass(
    const float* __restrict__ S, const float* __restrict__ v,
    float* __restrict__ u, float* __restrict__ pi) {
  extern __shared__ float smem[];
  float* tile   = smem;                       // 16 * 2052
  float* vsh    = tile + 16 * ROW_STRIDE;     // 2048
  float* pmax   = vsh + NN;                   // 256
  float* rowmax = pmax + 256;                 // 16
  float* psum   = rowmax + 16;                // 8 * 16
  float* urow   = psum + 128;                 // 16

  const int b    = blockIdx.y;
  const int row0 = blockIdx.x * 16;
  const int t    = threadIdx.x;
  const size_t sbase   = ((size_t)b * NN + row0) * NN;
  const uint32_t lds_base = (uint32_t)(uintptr_t)smem;  // flat LDS addr low 32 = LDS offset

  // ---- Stage v vector (async, b128 per lane) ----
  if (!V_ZERO) {
    for (int i = t; i < NN / 4; i += 256) {
      uint32_t loff = lds_base + (uint32_t)(16 * ROW_STRIDE + i * 4) * 4u;
      uint64_t gp = (uint64_t)(uintptr_t)(v + (size_t)b * NN + i * 4);
      asm volatile("global_load_async_to_lds_b128 %0, %1, off"
                   :: "v"(loff), "v"(gp) : "memory");
    }
  }
  // ---- Stage 16x2048 tile with +4-float row pad (async, b128 per lane) ----
  for (int i = t; i < 16 * (NN / 4); i += 256) {
    int r  = i >> 9;        // 512 float4 per row
    int c4 = i & 511;
    uint32_t loff = lds_base + (uint32_t)(r * ROW_STRIDE + c4 * 4) * 4u;
    uint64_t gp = (uint64_t)(uintptr_t)(S + sbase + (size_t)r * NN + c4 * 4);
    asm volatile("global_load_async_to_lds_b128 %0, %1, off"
                 :: "v"(loff), "v"(gp) : "memory");
  }
  asm volatile("s_wait_asynccnt 0" ::: "memory");
  __syncthreads();

  // ---- Row max: 16 threads per row ----
  const int r = t >> 4;
  const int j = t & 15;
  float m4 = -3.0e38f;
  for (int c4 = j; c4 < NN / 4; c4 += 16) {
    float4 x = *(const float4*)&tile[r * ROW_STRIDE + c4 * 4];
    if (!V_ZERO) {
      float4 vv = *(const float4*)&vsh[c4 * 4];
      x.x += vv.x; x.y += vv.y; x.z += vv.z; x.w += vv.w;
    }
    m4 = fmaxf(m4, fmaxf(fmaxf(x.x, x.y), fmaxf(x.z, x.w)));
  }
  pmax[r * 16 + j] = m4;
  __syncthreads();
  if (t < 16) {
    float m = pmax[t * 16];
    for (int k = 1; k < 16; ++k) m = fmaxf(m, pmax[t * 16 + k]);
    rowmax[t] = m;
  }
  __syncthreads();

  // ---- Row sum of exp via WMMA f32 16x16x4 with B = ones ----
  // A layout (ISA 7.12.2, 32-bit A 16x4): lane m (and m+16) holds row m;
  // VGPR0 = K=0 (lanes 0-15) / K=2 (lanes 16-31); VGPR1 = K=1 / K=3.
  const int wave = t >> 5;
  const int lane = t & 31;
  const int am   = lane & 15;
  const int hi   = (lane >> 4) << 1;  // 0 or 2
  const float rm = rowmax[am];
  v8f acc = {0.f, 0.f, 0.f, 0.f, 0.f, 0.f, 0.f, 0.f};
  const v2f ones = {1.0f, 1.0f};
  const int k0 = wave * (NN / 8);     // 256 cols per wave
  for (int k = k0; k < k0 + NN / 8; k += 4) {
    v2f x = *(const v2f*)&tile[am * ROW_STRIDE + k + hi];
    if (!V_ZERO) {
      v2f vv = *(const v2f*)&vsh[k + hi];
      x += vv;
    }
    v2f a;
    a.x = __expf(x.x - rm);
    a.y = __expf(x.y - rm);
    acc = __builtin_amdgcn_wmma_f32_16x16x4_f32(false, a, false, ones,
                                                (short)0, acc, false, false);
  }
  // D layout: VGPR q, lane 0 -> (M=q, N=0); lane 16 -> (M=q+8, N=0)
  if (lane == 0) {
#pragma unroll
    for (int q = 0; q < 8; ++q) psum[wave * 16 + q] = acc[q];
  } else if (lane == 16) {
#pragma unroll
    for (int q = 0; q < 8; ++q) psum[wave * 16 + 8 + q] = acc[q];
  }
  __syncthreads();

  // ---- Combine 8 wave partials -> u ----
  if (t < 16) {
    float s = 0.f;
    for (int w = 0; w < 8; ++w) s += psum[w * 16 + t];
    float uu = -LOG_N - (rowmax[t] + __logf(s));
    if (WRITE_PI) urow[t] = uu;
    else u[(size_t)b * NN + row0 + t] = uu;
  }

  if (WRITE_PI) {
    __syncthreads();
    const float ur = urow[r];
    for (int c4 = j; c4 < NN / 4; c4 += 16) {
      float4 x = *(const float4*)&tile[r * ROW_STRIDE + c4 * 4];
      float4 vv = make_float4(0.f, 0.f, 0.f, 0.f);
      if (!V_ZERO) vv = *(const float4*)&vsh[c4 * 4];
      float4 o;
      o.x = __expf(x.x + vv.x + ur);
      o.y = __expf(x.y + vv.y + ur);
      o.z = __expf(x.z + vv.z + ur);
      o.w = __expf(x.w + vv.w + ur);
      *(float4*)&pi[sbase + (size_t)r * NN + c4 * 4] = o;
    }
  }
}

// ---------------------------------------------------------------------------
// Column-LSE partial: block = 256 cols x 128 rows of one batch, fully
// coalesced, online (max, rescaled-sum) in chunks of 8 rows (~1.1 exp/elem).
// ---------------------------------------------------------------------------
__global__ __launch_bounds__(256) void sinkhorn_col_partial(
    const float* __restrict__ S, const float* __restrict__ u,
    float* __restrict__ pm, float* __restrict__ ps) {
  const int b  = blockIdx.z;
  const int cb = blockIdx.x;   // 8 column blocks
  const int rb = blockIdx.y;   // 16 row blocks
  const int c  = cb * 256 + threadIdx.x;
  const int r0 = rb * 128;
  const float* Sp = S + ((size_t)b * NN + r0) * NN + c;
  const float* up = u + (size_t)b * NN + r0;
  float m = -3.0e38f, s = 0.f;
  for (int chunk = 0; chunk < 128; chunk += 8) {
    float x[8];
#pragma unroll
    for (int jj = 0; jj < 8; ++jj)
      x[jj] = Sp[(size_t)(chunk + jj) * NN] + up[chunk + jj];
    float cm = x[0];
#pragma unroll
    for (int jj = 1; jj < 8; ++jj) cm = fmaxf(cm, x[jj]);
    float nm = fmaxf(m, cm);
    float acc = 0.f;
#pragma unroll
    for (int jj = 0; jj < 8; ++jj) acc += __expf(x[jj] - nm);
    s = s * __expf(m - nm) + acc;
    m = nm;
  }
  const size_t pidx = ((size_t)b * 16 + rb) * NN + c;
  pm[pidx] = m;
  ps[pidx] = s;
}

__global__ __launch_bounds__(256) void sinkhorn_col_combine(
    const float* __restrict__ pm, const float* __restrict__ ps,
    float* __restrict__ v) {
  const int idx = blockIdx.x * 256 + threadIdx.x;  // b*2048 + c
  const int b = idx >> 11, c = idx & (NN - 1);
  float M = -3.0e38f;
  for (int i = 0; i < 16; ++i)
    M = fmaxf(M, pm[((size_t)b * 16 + i) * NN + c]);
  float s = 0.f;
  for (int i = 0; i < 16; ++i) {
    const size_t p = ((size_t)b * 16 + i) * NN + c;
    s += ps[p] * __expf(pm[p] - M);
  }
  float vn = -LOG_N - (M + __logf(s));
  v[idx] = (vn > 9.0e8f) ? 0.f : vn;
}

// ---------------------------------------------------------------------------
extern "C" void kernel_launch(void* const* d_in, const int* in_sizes, int n_in,
                              void* d_out, int out_size, void* d_ws, size_t ws_size,
                              hipStream_t stream) {
  (void)in_sizes; (void)n_in; (void)out_size; (void)ws_size;
  const float* S = (const float*)d_in[0];
  float* pi = (float*)d_out;
  float* u  = (float*)d_ws;              // 16*2048
  float* v  = u + NB * NN;               // 16*2048
  float* pm = v + NB * NN;               // 16*16*2048
  float* ps = pm + NB * 16 * NN;         // 16*16*2048

  const size_t shmem =
      (size_t)(16 * ROW_STRIDE + NN + 256 + 16 + 128 + 16) * sizeof(float);
  hipFuncSetAttribute((const void*)sinkhorn_row_pass<true, false>,
                      hipFuncAttributeMaxDynamicSharedMemorySize, (int)shmem);
  hipFuncSetAttribute((const void*)sinkhorn_row_pass<false, false>,
                      hipFuncAttributeMaxDynamicSharedMemorySize, (int)shmem);
  hipFuncSetAttribute((const void*)sinkhorn_row_pass<false, true>,
                      hipFuncAttributeMaxDynamicSharedMemorySize, (int)shmem);

  dim3 rgrid(NN / 16, NB);          // 128 x 16 = 2048 blocks
  dim3 cgrid(NN / 256, 16, NB);     // 8 x 16 x 16 = 2048 blocks

  // i=0: u1 = f(v=0)
  sinkhorn_row_pass<true, false><<<rgrid, 256, shmem, stream>>>(S, v, u, nullptr);
  // i=1: v1 = g(u1)
  sinkhorn_col_partial<<<cgrid, 256, 0, stream>>>(S, u, pm, ps);
  sinkhorn_col_combine<<<NB * NN / 256, 256, 0, stream>>>(pm, ps, v);
  // i=2: u2 = f(v1)
  sinkhorn_row_pass<false, false><<<rgrid, 256, shmem, stream>>>(S, v, u, nullptr);
  // i=3: v2 = g(u2)
  sinkhorn_col_partial<<<cgrid, 256, 0, stream>>>(S, u, pm, ps);
  sinkhorn_col_combine<<<NB * NN / 256, 256, 0, stream>>>(pm, ps, v);
  // i=4 fused with output: u3 = f(v2), pi = exp(S + u3 + v2)
  sinkhorn_row_pass<false, true><<<rgrid, 256, shmem, stream>>>(S, v, u, pi);
}